// DictionaryLearning_16956530885037
// MI455X (gfx1250) — compile-verified
//
#include <hip/hip_runtime.h>
#include <hip/hip_bf16.h>

// ---------------------------------------------------------------------------
// DictionaryLearning (batch OMP sparse coding, VQ-style straight-through)
// MI455X / gfx1250: f32 WMMA (V_WMMA_F32_16X16X4_F32) for both GEMMs,
// per-wave register-resident OMP, deterministic loss reduction.
// ---------------------------------------------------------------------------

typedef float v2f __attribute__((ext_vector_type(2)));
typedef float v8f __attribute__((ext_vector_type(8)));

#define C_DIM   64
#define K_ATOMS 512
#define N_SIG   65536
#define SPARS   5

// workspace layout (in floats)
#define WS_G      0            // 512*512            = 262144
#define WS_DPACK  262144       // 32*32*16*2         =  32768
#define WS_SIDEI  294912       // int[N*5]           = 327680
#define WS_SIDEV  622592       // float[N*5]         = 327680
#define WS_PART   950272       // 16384 block partials

__device__ __forceinline__ v8f wmma4(v2f a, v2f b, v8f c) {
  // D = A(16x4,f32) * B(4x16,f32) + C(16x16,f32)
  return __builtin_amdgcn_wmma_f32_16x16x4_f32(
      /*neg_a=*/false, a, /*neg_b=*/false, b,
      /*c_mod=*/(short)0, c, /*reuse_a=*/false, /*reuse_b=*/false);
}

// ------------------------------ G = D^T D ----------------------------------
// grid (32,32) x 32 threads: one wave per 16x16 tile of G, K = 64 (16 steps).
__global__ __launch_bounds__(32)
void k_gram(const float* __restrict__ D, float* __restrict__ G) {
  const int lane = threadIdx.x;
  const int hf = lane >> 4, l = lane & 15;
  const int i0 = blockIdx.x * 16, j0 = blockIdx.y * 16;
  v8f acc = {0.f, 0.f, 0.f, 0.f, 0.f, 0.f, 0.f, 0.f};
#pragma unroll
  for (int q = 0; q < 16; ++q) {
    const int c = q * 4 + hf * 2;          // A: lanes 0-15 -> K0/K1, 16-31 -> K2/K3
    v2f a, b;
    a.x = D[c * K_ATOMS + i0 + l];
    a.y = D[(c + 1) * K_ATOMS + i0 + l];
    b.x = D[c * K_ATOMS + j0 + l];
    b.y = D[(c + 1) * K_ATOMS + j0 + l];
    acc = wmma4(a, b, acc);
  }
#pragma unroll
  for (int v = 0; v < 8; ++v) {            // C: vgpr v -> rows v / v+8
    const int M = v + hf * 8;
    G[(i0 + M) * K_ATOMS + j0 + l] = acc[v];
  }
}

// --------------- repack D into per-lane-contiguous B fragments -------------
// P[((ktile*32 + lane)*16 + q)*2 + e] = D[q*4 + (lane>>4)*2 + e][ktile*16 + (lane&15)]
__global__ __launch_bounds__(32)
void k_pack(const float* __restrict__ D, float* __restrict__ P) {
  const int kt = blockIdx.x, lane = threadIdx.x;
  const int hf = lane >> 4, l = lane & 15;
#pragma unroll
  for (int q = 0; q < 16; ++q) {
#pragma unroll
    for (int e = 0; e < 2; ++e) {
      const int c = q * 4 + hf * 2 + e;
      P[((kt * 32 + lane) * 16 + q) * 2 + e] = D[c * K_ATOMS + kt * 16 + l];
    }
  }
}

// ------------------------------ main OMP kernel ----------------------------
// 1024 blocks x 128 threads (4 waves). Wave w handles 16 consecutive signals.
// Dynamic LDS: init_corr[4][16][512] f32 + selI[64][5] + selV[64][5].
__global__ __launch_bounds__(128)
void k_omp(const float* __restrict__ z_e, const float* __restrict__ G,
           const float* __restrict__ P, float* __restrict__ coeffs,
           int* __restrict__ sideI, float* __restrict__ sideV) {
  extern __shared__ float lds[];
  float* corrL = lds;                          // 32768 floats
  int*   selI  = (int*)(lds + 32768);          // 320 ints
  float* selV  = lds + 32768 + 320;            // 320 floats

  const int tid = threadIdx.x;
  const int lane = tid & 31, wid = tid >> 5;
  const int hf = lane >> 4, l = lane & 15;
  const int n0wg = blockIdx.x * 64;
  const int n0 = n0wg + wid * 16;
  // zf[r][n] = z_e[r*65536 + (n&63)*1024 + ((n>>11)&31)*32 + ((n>>6)&31)]
  const int off = (((n0 >> 11) & 31) << 5) + ((n0 >> 6) & 31);

  // ---- load A fragments once: 16 signals x 64 channels ----
  v2f afr[16];
#pragma unroll
  for (int q = 0; q < 16; ++q) {
    const float* pa = z_e + (size_t)(q * 4 + hf * 2) * 65536
                          + (wid * 16 + l) * 1024 + off;
    afr[q].x = pa[0];
    afr[q].y = pa[65536];
  }

  float* myCorr = corrL + wid * (16 * K_ATOMS);

  // ---- init_corr GEMM: 32 k-tiles x 16 WMMAs, results to LDS ----
  for (int kt = 0; kt < 32; ++kt) {
    const float4* bp = (const float4*)(P + (size_t)(kt * 32 + lane) * 32);
    float4 bb[8];
#pragma unroll
    for (int t = 0; t < 8; ++t) bb[t] = bp[t];
    v8f acc = {0.f, 0.f, 0.f, 0.f, 0.f, 0.f, 0.f, 0.f};
#pragma unroll
    for (int q = 0; q < 16; ++q) {
      const float4 f = bb[q >> 1];
      v2f b;
      b.x = (q & 1) ? f.z : f.x;
      b.y = (q & 1) ? f.w : f.y;
      acc = wmma4(afr[q], b, acc);
    }
#pragma unroll
    for (int v = 0; v < 8; ++v)
      myCorr[(v + hf * 8) * K_ATOMS + kt * 16 + l] = acc[v];
  }

  // ---- per-signal OMP (corr row in registers: atom = s*32 + lane) ----
  for (int i = 0; i < 16; ++i) {
    float ic[16], cr[16];
#pragma unroll
    for (int s = 0; s < 16; ++s) {
      ic[s] = myCorr[i * K_ATOMS + s * 32 + lane];
      cr[s] = ic[s];
    }
    unsigned mask = 0xFFFFu;
    int   I[SPARS];
    float xs[SPARS];
    float Ld[15];                            // lower-tri 5x5, row-major packed

#pragma unroll
    for (int k = 1; k <= SPARS; ++k) {
      // masked abs-argmax over 512 atoms (ties -> lowest index, as jnp.argmax)
      float bv = -1.f;
      int ba = 0x7FFFFFFF;
#pragma unroll
      for (int s = 0; s < 16; ++s) {
        const float v = ((mask >> s) & 1u) ? fabsf(cr[s]) : -1.f;
        const int aidx = s * 32 + lane;
        if (v > bv || (v == bv && aidx < ba)) { bv = v; ba = aidx; }
      }
#pragma unroll
      for (int o = 16; o > 0; o >>= 1) {
        const float ov = __shfl_xor(bv, o, 32);
        const int   oa = __shfl_xor(ba, o, 32);
        if (ov > bv || (ov == bv && oa < ba)) { bv = ov; ba = oa; }
      }
      const int kh = __builtin_amdgcn_readfirstlane(ba);
      if (lane == (kh & 31)) mask &= ~(1u << (kh >> 5));
      __builtin_prefetch(&G[(size_t)kh * K_ATOMS + lane], 0, 1);

      // progressive Cholesky update (wave-uniform scalar math)
      if (k == 1) {
        Ld[0] = 1.f;
      } else {
        const int m = k - 1;
        float w[4];
#pragma unroll
        for (int j = 0; j < 4; ++j) {
          if (j < m) {
            float g = G[(size_t)I[j] * K_ATOMS + kh];
#pragma unroll
            for (int t = 0; t < 4; ++t)
              if (t < j) g -= Ld[j * (j + 1) / 2 + t] * w[t];
            w[j] = g / Ld[j * (j + 1) / 2 + j];
          }
        }
        float s2 = 0.f;
#pragma unroll
        for (int j = 0; j < 4; ++j)
          if (j < m) s2 += w[j] * w[j];
        const float wbr = sqrtf(1.f - s2);
#pragma unroll
        for (int t = 0; t < 4; ++t)
          if (t < m) Ld[m * (m + 1) / 2 + t] = w[t];
        Ld[m * (m + 1) / 2 + m] = wbr;
      }
      I[k - 1] = kh;

      // b = init_corr[I] (uniform LDS reads); solve L y = b; L^T x = y
      float bvec[SPARS], y[SPARS], x[SPARS];
#pragma unroll
      for (int j = 0; j < SPARS; ++j)
        if (j < k) bvec[j] = myCorr[i * K_ATOMS + I[j]];
#pragma unroll
      for (int j = 0; j < SPARS; ++j) {
        if (j < k) {
          float t0 = bvec[j];
#pragma unroll
          for (int t = 0; t < SPARS; ++t)
            if (t < j) t0 -= Ld[j * (j + 1) / 2 + t] * y[t];
          y[j] = t0 / Ld[j * (j + 1) / 2 + j];
        }
      }
#pragma unroll
      for (int jj = SPARS - 1; jj >= 0; --jj) {
        if (jj < k) {
          float t0 = y[jj];
#pragma unroll
          for (int t = 0; t < SPARS; ++t)
            if (t > jj && t < k) t0 -= Ld[t * (t + 1) / 2 + jj] * x[t];
          x[jj] = t0 / Ld[jj * (jj + 1) / 2 + jj];
        }
      }
#pragma unroll
      for (int j = 0; j < SPARS; ++j)
        if (j < k) xs[j] = x[j];

      // corr = init_corr - G[I]^T xs  (skip after final iteration)
      if (k < SPARS) {
#pragma unroll
        for (int s = 0; s < 16; ++s) {
          float g = 0.f;
#pragma unroll
          for (int j = 0; j < SPARS; ++j)
            if (j < k) g += xs[j] * G[(size_t)I[j] * K_ATOMS + s * 32 + lane];
          cr[s] = ic[s] - g;
        }
      }
    }

    // record selection (wave-uniform; lane 0 writes)
    const int n = n0 + i;
    if (lane == 0) {
#pragma unroll
      for (int j = 0; j < SPARS; ++j) {
        selI[(wid * 16 + i) * SPARS + j] = I[j];
        selV[(wid * 16 + i) * SPARS + j] = xs[j];
        sideI[(size_t)n * SPARS + j] = I[j];
        sideV[(size_t)n * SPARS + j] = xs[j];
      }
    }
  }

  __syncthreads();

  // ---- dense coeffs block write: [512][64] for this WG's 64 columns ----
  for (int it = 0; it < 64; ++it) {
    const int flat4 = it * 128 + tid;
    const int k = flat4 >> 4;                // atom row 0..511
    const int c4 = flat4 & 15;               // 4-column group 0..15
    float* dst = coeffs + (size_t)k * N_SIG + n0wg + c4 * 4;
#pragma unroll
    for (int e = 0; e < 4; ++e) {
      const int lc = c4 * 4 + e;
      float v = 0.f;
#pragma unroll
      for (int j = 0; j < SPARS; ++j)
        if (selI[lc * SPARS + j] == k) v = selV[lc * SPARS + j];
      dst[e] = v;
    }
  }
}

// -------------------- z_st (BCHW) + per-block loss partials ----------------
__global__ __launch_bounds__(256)
void k_out(const float* __restrict__ z_e, const float* __restrict__ D,
           const int* __restrict__ sideI, const float* __restrict__ sideV,
           float* __restrict__ z_out, float* __restrict__ partials) {
  const int idx = blockIdx.x * 256 + threadIdx.x;       // BCHW flat index
  const int w = idx & 31, h = (idx >> 5) & 31;
  const int cc = (idx >> 10) & 63, b = idx >> 16;
  const int n = h * 2048 + w * 64 + cc;                 // column of [64,65536]
  float acc = 0.f;
#pragma unroll
  for (int j = 0; j < SPARS; ++j) {
    const int a = sideI[(size_t)n * SPARS + j];
    acc += sideV[(size_t)n * SPARS + j] * D[b * K_ATOMS + a];
  }
  z_out[idx] = acc;                                     // z_st forward == z_dl
  const float d = acc - z_e[idx];                       // z(BHWC)[b,h,w,cc] == z_e[idx]
  float p = d * d;
#pragma unroll
  for (int o = 16; o; o >>= 1) p += __shfl_xor(p, o, 32);
  __shared__ float red[8];
  if ((threadIdx.x & 31) == 0) red[threadIdx.x >> 5] = p;
  __syncthreads();
  if (threadIdx.x == 0) {
    float s = 0.f;
#pragma unroll
    for (int t = 0; t < 8; ++t) s += red[t];
    partials[blockIdx.x] = s;
  }
}

// ---------------------- deterministic loss reduction -----------------------
__global__ __launch_bounds__(256)
void k_loss(const float* __restrict__ partials, float* __restrict__ loss) {
  float s = 0.f;
  for (int t = threadIdx.x; t < 16384; t += 256) s += partials[t];
#pragma unroll
  for (int o = 16; o; o >>= 1) s += __shfl_xor(s, o, 32);
  __shared__ float red[8];
  if ((threadIdx.x & 31) == 0) red[threadIdx.x >> 5] = s;
  __syncthreads();
  if (threadIdx.x == 0) {
    float t = 0.f;
#pragma unroll
    for (int q = 0; q < 8; ++q) t += red[q];
    // loss = 0.25*mse + mse = 1.25 * sum(d^2) / (64*64*32*32)
    *loss = t * (1.25f / 4194304.0f);
  }
}

// ---------------------------------------------------------------------------
extern "C" void kernel_launch(void* const* d_in, const int* in_sizes, int n_in,
                              void* d_out, int out_size, void* d_ws, size_t ws_size,
                              hipStream_t stream) {
  const float* z_e = (const float*)d_in[0];       // [64,64,32,32] f32
  const float* D   = (const float*)d_in[1];       // [64,512] f32 (pre-normalized)

  float* ws = (float*)d_ws;
  float* G     = ws + WS_G;
  float* P     = ws + WS_DPACK;
  int*   sI    = (int*)(ws + WS_SIDEI);
  float* sV    = ws + WS_SIDEV;
  float* part  = ws + WS_PART;

  float* out    = (float*)d_out;
  float* z_out  = out;                            // 4194304 floats (BCHW)
  float* loss   = out + 4194304;                  // 1 float
  float* coeffs = out + 4194305;                  // 512*65536 floats

  k_gram<<<dim3(32, 32), 32, 0, stream>>>(D, G);
  k_pack<<<32, 32, 0, stream>>>(D, P);

  const size_t ldsz = (size_t)(32768 + 640) * sizeof(float);  // 130.5 KB
  k_omp<<<1024, 128, ldsz, stream>>>(z_e, G, P, coeffs, sI, sV);

  k_out<<<16384, 256, 0, stream>>>(z_e, D, sI, sV, z_out, part);
  k_loss<<<1, 256, 0, stream>>>(part, loss);
}